// MultiHeadMLP_39633958208207
// MI455X (gfx1250) — compile-verified
//
#include <hip/hip_runtime.h>
#include <hip/hip_bf16.h>

// ---------------------------------------------------------------------------
// GAT layer for MI455X (gfx1250, wave32).
// - f32 GEMMs via V_WMMA_F32_16X16X4_F32 (native f32 matrix op)
// - edge_attr streamed into LDS via GLOBAL_LOAD_ASYNC_TO_LDS_B128 (ASYNCcnt)
// - global_prefetch_b8 lookahead on the edge stream
// - softmax via float-ordered atomic max + f32 atomic adds (node arrays are
//   L2-resident: 192 MB L2 >> all node-side state)
// ---------------------------------------------------------------------------

typedef float v2f __attribute__((ext_vector_type(2)));
typedef float v4f __attribute__((ext_vector_type(4)));
typedef float v8f __attribute__((ext_vector_type(8)));

#define GN 100000   // nodes
#define GE 1600000  // edges
#define GD 64
#define GH 8

// CDNA5 async global->LDS staging (inline asm per ISA §15.18.3, opc 98).
// Flip to 0 to fall back to synchronous VGPR round-trip staging.
#define USE_ASYNC_LDS 1

__device__ __forceinline__ void atomicMaxF32(float* addr, float val) {
  // Monotonic float ordering trick: works for mixed signs with -FLT_MAX init.
  if (val >= 0.0f) atomicMax((int*)addr, __float_as_int(val));
  else             atomicMin((unsigned int*)addr, __float_as_uint(val));
}

#if USE_ASYNC_LDS
// LDS byte offset of a generic pointer that points into shared memory.
__device__ __forceinline__ unsigned lds_off(const void* p) {
  return (unsigned)(unsigned long long)(uintptr_t)
         (__attribute__((address_space(3))) const void*)p;
}
#endif

// ---------------------------------------------------------------------------
// Y[nrows,64] = (X + preBias) @ W^T + colBias        (W is [64,64] row-major)
// One wave per 16-row tile; 4 column tiles of 16; K=64 as 16 steps of
// V_WMMA_F32_16X16X4_F32. A-fragments preloaded once and reused over jt.
// ---------------------------------------------------------------------------
__global__ void gemm64_wmma_kernel(const float* __restrict__ X,
                                   const float* __restrict__ W,
                                   const float* __restrict__ preBias,
                                   const float* __restrict__ colBias,
                                   float* __restrict__ Y, int nrows) {
  int gwave = (int)((blockIdx.x * blockDim.x + threadIdx.x) >> 5);
  int lane  = threadIdx.x & 31;
  int ntiles = nrows >> 4;
  if (gwave >= ntiles) return;

  int m0   = gwave << 4;
  int half = lane >> 4;      // which K-pair of the 4-wide K step
  int l16  = lane & 15;
  size_t row = (size_t)(m0 + l16);

  v2f afrag[16];
#pragma unroll
  for (int ks = 0; ks < 16; ++ks) {
    int kk = ks * 4 + half * 2;
    v2f a = *(const v2f*)&X[row * 64 + kk];
    if (preBias) {
      v2f pb = *(const v2f*)&preBias[kk];
      a.x += pb.x; a.y += pb.y;
    }
    afrag[ks] = a;
  }

  for (int jt = 0; jt < 4; ++jt) {
    int j = (jt << 4) + l16;
    v8f c = {};
#pragma unroll
    for (int ks = 0; ks < 16; ++ks) {
      int kk = ks * 4 + half * 2;
      v2f b = *(const v2f*)&W[(size_t)j * 64 + kk];
      c = __builtin_amdgcn_wmma_f32_16x16x4_f32(
          /*neg_a=*/false, afrag[ks], /*neg_b=*/false, b,
          /*c_mod=*/(short)0, c, /*reuse_a=*/false, /*reuse_b=*/false);
    }
    float cb = colBias[j];
#pragma unroll
    for (int g = 0; g < 8; ++g) {
      size_t ro = (size_t)(m0 + g + half * 8);
      Y[ro * 64 + j] = c[g] + cb;
    }
  }
}

// ---------------------------------------------------------------------------
// Per-node attention scores: Sc[n,h] = h[n]*a_c(h), Sr[n,h] = h[n]*a_r(h).
// att is (8 heads, 20, 1): [0:8]=a_c, [8:16]=a_r, [16:20]=a_e.
// ---------------------------------------------------------------------------
__global__ void node_scores_kernel(const float* __restrict__ Hf,
                                   const float* __restrict__ att,
                                   float* __restrict__ Sc,
                                   float* __restrict__ Sr) {
  int n = blockIdx.x * blockDim.x + threadIdx.x;
  if (n >= GN) return;
  const float* hn = Hf + (size_t)n * 64;
#pragma unroll
  for (int hh = 0; hh < 8; ++hh) {
    float sc = 0.0f, sr = 0.0f;
#pragma unroll
    for (int d = 0; d < 8; ++d) {
      float v = hn[hh * 8 + d];
      sc = fmaf(v, att[hh * 20 + d], sc);
      sr = fmaf(v, att[hh * 20 + 8 + d], sr);
    }
    Sc[(size_t)n * 8 + hh] = sc;
    Sr[(size_t)n * 8 + hh] = sr;
  }
}

// ---------------------------------------------------------------------------
// Me[k,h] = sum_{j in head h's 4-block} W_edge[j,k] * a_e[h, j-4h]   (32x8)
// Folds the edge GEMM into the attention dot: alpha_e = edge_attr @ Me.
// ---------------------------------------------------------------------------
__global__ void build_me_kernel(const float* __restrict__ Wedge,
                                const float* __restrict__ att,
                                float* __restrict__ Me) {
  int t = threadIdx.x;          // 256 threads: (k, h)
  int k = t >> 3, hh = t & 7;
  float s = 0.0f;
#pragma unroll
  for (int jj = 0; jj < 4; ++jj) {
    int j = hh * 4 + jj;
    s = fmaf(Wedge[(size_t)j * 32 + k], att[hh * 20 + 16 + jj], s);
  }
  Me[k * 8 + hh] = s;
}

__global__ void init_ws_kernel(float* __restrict__ amax,
                               float* __restrict__ den,
                               float* __restrict__ agg) {
  long i = (long)blockIdx.x * blockDim.x + threadIdx.x;
  if (i < (long)GN * 8) { amax[i] = -3.402823466e38f; den[i] = 0.0f; }
  if (i < (long)GN * 64) agg[i] = 0.0f;
}

// ---------------------------------------------------------------------------
// Edge pass 1: stage 32 edges x 32 feats per wave into LDS with async
// global->LDS b128 copies (coalesced 512B per instruction per wave, padded
// stride 36 floats), then alpha = Sc[c]+Sr[r]+ea@Me, leaky-relu, store to
// alphaE, atomic-max into amax[c].  E divisible by 256: no guards.
// ---------------------------------------------------------------------------
__global__ void edge_alpha_kernel(const int* __restrict__ ei,
                                  const float* __restrict__ ea,
                                  const float* __restrict__ Me,
                                  const float* __restrict__ Sc,
                                  const float* __restrict__ Sr,
                                  float* __restrict__ alphaE,
                                  float* __restrict__ amax) {
  __shared__ float sMe[256];
  __shared__ __align__(16) float tile[8][32 * 36];

  sMe[threadIdx.x] = Me[threadIdx.x];

  int wid  = threadIdx.x >> 5;
  int lane = threadIdx.x & 31;
  long wbase = (long)blockIdx.x * 256 + (long)wid * 32;  // first edge of wave

  // Prefetch next block's chunk of the edge stream (one 128B line per thread).
  if ((int)blockIdx.x + 1 < (int)gridDim.x) {
    __builtin_prefetch(&ea[((size_t)blockIdx.x + 1) * 256 * 32 +
                           (size_t)threadIdx.x * 32], 0, 1);
  }

#pragma unroll
  for (int rep = 0; rep < 8; ++rep) {
    int idx  = rep * 128 + lane * 4;     // flat float index in 1024-float span
    int eloc = idx >> 5;                 // local edge
    int feat = idx & 31;                 // feature
    const float* gsrc = &ea[(size_t)wbase * 32 + idx];
#if USE_ASYNC_LDS
    unsigned ldst = lds_off(&tile[wid][eloc * 36 + feat]);
    asm volatile("global_load_async_to_lds_b128 %0, %1, off"
                 :: "v"(ldst),
                    "v"((unsigned long long)(uintptr_t)gsrc)
                 : "memory");
#else
    *(v4f*)&tile[wid][eloc * 36 + feat] = *(const v4f*)gsrc;
#endif
  }
#if USE_ASYNC_LDS
  asm volatile("s_wait_asynccnt 0" ::: "memory");
#endif
  __syncthreads();

  long e = wbase + lane;
  int r = ei[e * 2 + 0];
  int c = ei[e * 2 + 1];

  const float* mine = &tile[wid][lane * 36];
  float acc[8] = {0, 0, 0, 0, 0, 0, 0, 0};
#pragma unroll
  for (int kq = 0; kq < 8; ++kq) {
    v4f v = *(const v4f*)&mine[kq * 4];
#pragma unroll
    for (int u = 0; u < 4; ++u) {
      int k = kq * 4 + u;
      float x = v[u];
#pragma unroll
      for (int hh = 0; hh < 8; ++hh) acc[hh] = fmaf(x, sMe[k * 8 + hh], acc[hh]);
    }
  }

  float scv[8], srv[8];
  *(v4f*)&scv[0] = *(const v4f*)&Sc[(size_t)c * 8];
  *(v4f*)&scv[4] = *(const v4f*)&Sc[(size_t)c * 8 + 4];
  *(v4f*)&srv[0] = *(const v4f*)&Sr[(size_t)r * 8];
  *(v4f*)&srv[4] = *(const v4f*)&Sr[(size_t)r * 8 + 4];

  float aout[8];
#pragma unroll
  for (int hh = 0; hh < 8; ++hh) {
    float s = acc[hh] + scv[hh] + srv[hh];
    s = s > 0.0f ? s : 0.01f * s;       // leaky_relu(0.01)
    aout[hh] = s;
    atomicMaxF32(&amax[(size_t)c * 8 + hh], s);
  }
  *(v4f*)&alphaE[(size_t)e * 8]     = *(v4f*)&aout[0];
  *(v4f*)&alphaE[(size_t)e * 8 + 4] = *(v4f*)&aout[4];
}

// Edge pass 2: den[c,h] += exp(alpha - amax[c,h])
__global__ void edge_den_kernel(const int* __restrict__ ei,
                                const float* __restrict__ alphaE,
                                const float* __restrict__ amax,
                                float* __restrict__ den) {
  long e = (long)blockIdx.x * blockDim.x + threadIdx.x;
  if (e >= GE) return;
  int c = ei[e * 2 + 1];
  float a[8], m[8];
  *(v4f*)&a[0] = *(const v4f*)&alphaE[(size_t)e * 8];
  *(v4f*)&a[4] = *(const v4f*)&alphaE[(size_t)e * 8 + 4];
  *(v4f*)&m[0] = *(const v4f*)&amax[(size_t)c * 8];
  *(v4f*)&m[4] = *(const v4f*)&amax[(size_t)c * 8 + 4];
#pragma unroll
  for (int hh = 0; hh < 8; ++hh)
    atomicAdd(&den[(size_t)c * 8 + hh], __expf(a[hh] - m[hh]));
}

// Edge pass 3: agg[c,:] += (exp(alpha-amax)/(den+eps)) * h[r,:]
__global__ void edge_msg_kernel(const int* __restrict__ ei,
                                const float* __restrict__ alphaE,
                                const float* __restrict__ amax,
                                const float* __restrict__ den,
                                const float* __restrict__ Hf,
                                float* __restrict__ agg) {
  long e = (long)blockIdx.x * blockDim.x + threadIdx.x;
  if (e >= GE) return;
  int r = ei[e * 2 + 0];
  int c = ei[e * 2 + 1];
  float a[8], m[8], dn[8], w[8];
  *(v4f*)&a[0]  = *(const v4f*)&alphaE[(size_t)e * 8];
  *(v4f*)&a[4]  = *(const v4f*)&alphaE[(size_t)e * 8 + 4];
  *(v4f*)&m[0]  = *(const v4f*)&amax[(size_t)c * 8];
  *(v4f*)&m[4]  = *(const v4f*)&amax[(size_t)c * 8 + 4];
  *(v4f*)&dn[0] = *(const v4f*)&den[(size_t)c * 8];
  *(v4f*)&dn[4] = *(const v4f*)&den[(size_t)c * 8 + 4];
#pragma unroll
  for (int hh = 0; hh < 8; ++hh)
    w[hh] = __expf(a[hh] - m[hh]) / (dn[hh] + 1e-16f);

  const float* hr = Hf + (size_t)r * 64;
  float* ag = agg + (size_t)c * 64;
#pragma unroll
  for (int hh = 0; hh < 8; ++hh) {
    v4f h0 = *(const v4f*)&hr[hh * 8];
    v4f h1 = *(const v4f*)&hr[hh * 8 + 4];
    atomicAdd(&ag[hh * 8 + 0], w[hh] * h0[0]);
    atomicAdd(&ag[hh * 8 + 1], w[hh] * h0[1]);
    atomicAdd(&ag[hh * 8 + 2], w[hh] * h0[2]);
    atomicAdd(&ag[hh * 8 + 3], w[hh] * h0[3]);
    atomicAdd(&ag[hh * 8 + 4], w[hh] * h1[0]);
    atomicAdd(&ag[hh * 8 + 5], w[hh] * h1[1]);
    atomicAdd(&ag[hh * 8 + 6], w[hh] * h1[2]);
    atomicAdd(&ag[hh * 8 + 7], w[hh] * h1[3]);
  }
}

// ---------------------------------------------------------------------------
extern "C" void kernel_launch(void* const* d_in, const int* in_sizes, int n_in,
                              void* d_out, int out_size, void* d_ws, size_t ws_size,
                              hipStream_t stream) {
  (void)in_sizes; (void)n_in; (void)out_size; (void)ws_size;

  const float* feats  = (const float*)d_in[0];
  const int*   ei     = (const int*)  d_in[1];
  const float* ea     = (const float*)d_in[2];
  const float* W_fc   = (const float*)d_in[3];
  const float* b_fc   = (const float*)d_in[4];
  const float* W_edge = (const float*)d_in[5];
  const float* att    = (const float*)d_in[6];
  const float* bias   = (const float*)d_in[7];
  const float* W_out  = (const float*)d_in[8];
  const float* b_out  = (const float*)d_in[9];
  float* out = (float*)d_out;
  float* ws  = (float*)d_ws;

  // Workspace layout (floats); total ~28.81M floats (~115 MB).
  float* Hf     = ws;                       // N*64   = 6,400,000
  float* Sc     = Hf     + (size_t)GN * 64; // N*8    =   800,000
  float* Sr     = Sc     + (size_t)GN * 8;  // N*8
  float* Me     = Sr     + (size_t)GN * 8;  // 256
  float* alphaE = Me     + 256;             // E*8    = 12,800,000
  float* amax   = alphaE + (size_t)GE * 8;  // N*8
  float* den    = amax   + (size_t)GN * 8;  // N*8
  float* agg    = den    + (size_t)GN * 8;  // N*64

  const int ntiles = GN / 16;  // 6250 (exact)

  // 1) h = feats @ W_fc^T + b_fc   (WMMA f32)
  gemm64_wmma_kernel<<<(ntiles + 7) / 8, 256, 0, stream>>>(
      feats, W_fc, nullptr, b_fc, Hf, GN);

  // 2) per-node attention scores
  node_scores_kernel<<<(GN + 255) / 256, 256, 0, stream>>>(Hf, att, Sc, Sr);

  // 3) folded edge matrix Me = W_edge^T @ Ae  (32x8)
  build_me_kernel<<<1, 256, 0, stream>>>(W_edge, att, Me);

  // 4) init amax/den/agg
  init_ws_kernel<<<((long)GN * 64 + 255) / 256, 256, 0, stream>>>(amax, den, agg);

  // 5) edge logits + leaky relu + segment max   (E/256 = 6250 blocks exact)
  edge_alpha_kernel<<<GE / 256, 256, 0, stream>>>(ei, ea, Me, Sc, Sr, alphaE, amax);

  // 6) softmax denominator
  edge_den_kernel<<<GE / 256, 256, 0, stream>>>(ei, alphaE, amax, den);

  // 7) weighted message scatter-add
  edge_msg_kernel<<<GE / 256, 256, 0, stream>>>(ei, alphaE, amax, den, Hf, agg);

  // 8) out = (agg + bias) @ W_out^T + b_out    (WMMA f32) -> tuple slot 0
  gemm64_wmma_kernel<<<(ntiles + 7) / 8, 256, 0, stream>>>(
      agg, W_out, bias, b_out, out, GN);

  // 9) tuple slots 1,2: raw copies of edge_index and edge_attr
  float* out_ei = out + (size_t)GN * 64;
  float* out_ea = out_ei + (size_t)GE * 2;
  hipMemcpyAsync(out_ei, ei, (size_t)GE * 2 * sizeof(int),
                 hipMemcpyDeviceToDevice, stream);
  hipMemcpyAsync(out_ea, ea, (size_t)GE * 32 * sizeof(float),
                 hipMemcpyDeviceToDevice, stream);
}